// Custom_Attention_37108517437506
// MI455X (gfx1250) — compile-verified
//
#include <hip/hip_runtime.h>
#include <hip/hip_bf16.h>

typedef float v2f __attribute__((ext_vector_type(2)));
typedef float v8f __attribute__((ext_vector_type(8)));

#define N_COLS 2048
#define B_ROWS 32
#define INV_SCALE 0.0625f               // 1/16
#define LOG2E     1.4426950408889634f   // log2(e)

// One wave (32 lanes) handles a 16-row i-tile for one batch row.
// Block = 128 threads = 4 waves = 4 consecutive i-tiles.
// Grid  = 32 batches * 32 tile-groups = 1024 blocks.
__global__ __launch_bounds__(128)
void rank1_attn_wmma_kernel(const float* __restrict__ x, float* __restrict__ y) {
    __shared__ float xs[N_COLS];

    const int b         = blockIdx.x >> 5;          // batch row 0..31
    const int tileGroup = blockIdx.x & 31;          // 0..31
    const int wave      = threadIdx.x >> 5;         // 0..3
    const int lane      = threadIdx.x & 31;
    const int half      = lane >> 4;                // 0: lanes 0-15, 1: lanes 16-31
    const int n         = lane & 15;                // column index within tile

    const float* xrow = x + b * N_COLS;

    // Stage the whole batch row in LDS (8 KB), float4-vectorized.
    {
        const float4* src = reinterpret_cast<const float4*>(xrow);
        float4*       dst = reinterpret_cast<float4*>(xs);
        for (int t = threadIdx.x; t < N_COLS / 4; t += 128) dst[t] = src[t];
    }
    __syncthreads();

    const int iBase = (tileGroup * 4 + wave) * 16;  // first i of this wave's tile

    // Per-lane constants:
    //  s2    = x_i * log2(e) / 16  ->  exp(x_i*x_j/16) == exp2(s2 * x_j)
    //          (folds __expf's log2e multiply into the loop-invariant constant,
    //           saving one VALU mul per element; v_exp_f32 is base-2 natively)
    //  selN0 = lane provides B column 0 (x_j);  cB = 1.0 for column 1, else 0
    const float s2    = xs[iBase + n] * (INV_SCALE * LOG2E);
    const bool  selN0 = (n == 0);
    const float cB    = (n == 1) ? 1.0f : 0.0f;

    // 4 independent accumulators to break the WMMA->WMMA RAW chain on C.
    v8f acc0 = {}, acc1 = {}, acc2 = {}, acc3 = {};

    // K-offset within each 4-wide j step for this lane:
    //  A layout: lanes 0-15 hold K=0,1 ; lanes 16-31 hold K=2,3
    //  B layout: VGPR v holds row K = v + 2*half  -> same pair of x_j values.
    const int kOff = 2 * half;

    for (int j = 0; j < N_COLS; j += 16) {
        #pragma unroll
        for (int u = 0; u < 4; ++u) {
            const int jj = j + 4 * u;
            // ds_load_b64: the two x_j values this lane needs for both A and B.
            float2 xv = *reinterpret_cast<const float2*>(&xs[jj + kOff]);

            // A tile elements: exp2(s2 * x_j) == exp(x_i*x_j/16)   (TRANS pipe)
            float a0 = __builtin_amdgcn_exp2f(s2 * xv.x);
            float a1 = __builtin_amdgcn_exp2f(s2 * xv.y);

            // B tile elements: col0 = x_j, col1 = 1.0, cols 2..15 = 0
            float b0 = selN0 ? xv.x : cB;
            float b1 = selN0 ? xv.y : cB;

            v2f A = {a0, a1};
            v2f B = {b0, b1};

            switch (u) {
            case 0: acc0 = __builtin_amdgcn_wmma_f32_16x16x4_f32(false, A, false, B, (short)0, acc0, false, false); break;
            case 1: acc1 = __builtin_amdgcn_wmma_f32_16x16x4_f32(false, A, false, B, (short)0, acc1, false, false); break;
            case 2: acc2 = __builtin_amdgcn_wmma_f32_16x16x4_f32(false, A, false, B, (short)0, acc2, false, false); break;
            case 3: acc3 = __builtin_amdgcn_wmma_f32_16x16x4_f32(false, A, false, B, (short)0, acc3, false, false); break;
            }
        }
    }

    // Combine accumulators (C/D layout is elementwise per lane/VGPR, so vector add is valid).
    v8f t = acc0 + acc1 + acc2 + acc3;

    // C/D layout: VGPR v, lane -> M = v + 8*half, N = lane%16.
    // num_i sits in N==0 lanes (0,16); den_i in N==1 lanes (1,17).
    float* yrow = y + b * N_COLS;
    #pragma unroll
    for (int v = 0; v < 8; ++v) {
        float num = __shfl(t[v], half * 16 + 0, 32);
        float den = __shfl(t[v], half * 16 + 1, 32);
        float yv  = num / den;
        if (n == v) {                       // 16 lanes store one row each
            yrow[iBase + 8 * half + v] = yv;
        }
    }
}

extern "C" void kernel_launch(void* const* d_in, const int* in_sizes, int n_in,
                              void* d_out, int out_size, void* d_ws, size_t ws_size,
                              hipStream_t stream) {
    (void)in_sizes; (void)n_in; (void)out_size; (void)d_ws; (void)ws_size;
    const float* x = (const float*)d_in[0];   // [32, 2048] f32
    float*       y = (float*)d_out;           // [32, 2048] f32

    dim3 grid(B_ROWS * (N_COLS / 16 / 4));    // 32 * 32 = 1024 blocks
    dim3 block(128);                          // 4 waves, one i-tile each
    rank1_attn_wmma_kernel<<<grid, block, 0, stream>>>(x, y);
}